// SAGELayer_5617817224169
// MI455X (gfx1250) — compile-verified
//
#include <hip/hip_runtime.h>

typedef __attribute__((ext_vector_type(2))) float v2f;
typedef __attribute__((ext_vector_type(8))) float v8f;

#define NN 100000   // nodes
#define NE 1250000  // edges
#define D  64       // D_IN == D_E == D_OUT
#define DK 128      // concat width

// Workspace layout (float):
//   S  : [NN][DK]  S[v][0:64] = sum over in-edges of nfeats[src], S[v][64:128] = sum efeats
//   cnt: [NN]      in-degree (float)
// Total = NN*(DK+1)*4 bytes = 51.6 MB

__global__ void zero_ws_kernel(float4* __restrict__ ws, int n4) {
  int i = blockIdx.x * blockDim.x + threadIdx.x;
  if (i < n4) ws[i] = make_float4(0.f, 0.f, 0.f, 0.f);
}

// 64 threads per edge: thread f handles feature f of both the gathered node row
// (L2-resident, 25.6MB) and the streamed edge row. Accumulate into S via L2 atomics.
__global__ __launch_bounds__(256) void edge_scatter_kernel(
    const float* __restrict__ nfeats, const float* __restrict__ efeats,
    const int* __restrict__ src, const int* __restrict__ dst,
    float* __restrict__ S, float* __restrict__ cnt) {
  long tid = (long)blockIdx.x * blockDim.x + threadIdx.x;
  long e = tid >> 6;
  int  f = (int)(tid & 63);
  if (e >= NE) return;
  int s = src[e];
  int d = dst[e];
  float hv = nfeats[(long)s * D + f];
  float ev = efeats[e * D + f];
  float* Sd = S + (long)d * DK;
  atomicAdd(Sd + f,     hv);
  atomicAdd(Sd + D + f, ev);
  if (f == 0) atomicAdd(cnt + d, 1.0f);
}

// One wave32 per 16-node tile; 8 waves (128 nodes) per block.
// Stage 1: T = W_msg @ S  (K=128) via v_wmma_f32_16x16x4_f32
//          h_neigh = (T + cnt*b_msg) / max(cnt,1)  -> LDS (layout transpose)
// Stage 2: out = relu(W_apply @ concat(nfeats, h_neigh) + b_apply)
__global__ __launch_bounds__(256) void node_wmma_kernel(
    const float* __restrict__ nfeats,
    const float* __restrict__ S, const float* __restrict__ cnt,
    const float* __restrict__ Wmsg, const float* __restrict__ bmsg,
    const float* __restrict__ Wapp, const float* __restrict__ bapp,
    float* __restrict__ out) {
  __shared__ float hstage[8][16][D];  // 32 KB: per-wave h_neigh tile

  const int lane = threadIdx.x & 31;
  const int wave = threadIdx.x >> 5;
  const int tile = blockIdx.x * 8 + wave;
  const int node0 = tile * 16;              // NN % 16 == 0 -> full tiles only
  const bool active = (node0 < NN);         // uniform per wave => EXEC all-ones

  const int m   = lane & 15;                // A row / B col / D col
  const int kb  = (lane >> 4) << 1;         // K sub-offset {0,2} within a k-step
  const int rhi = (lane >> 4) << 3;         // D row offset {0,8}

  v8f acc0 = {}, acc1 = {}, acc2 = {}, acc3 = {};

  if (active) {
    // ---- Stage 1: [16x128] @ W_msg^T[128x64] ----
    const float* Xrow = S + (long)(node0 + m) * DK;
    const float* W0 = Wmsg + (long)(m)      * DK;
    const float* W1 = Wmsg + (long)(m + 16) * DK;
    const float* W2 = Wmsg + (long)(m + 32) * DK;
    const float* W3 = Wmsg + (long)(m + 48) * DK;
#pragma unroll 8
    for (int k0 = 0; k0 < DK; k0 += 4) {
      v2f a  = *(const v2f*)(Xrow + k0 + kb);
      v2f b0 = *(const v2f*)(W0 + k0 + kb);
      v2f b1 = *(const v2f*)(W1 + k0 + kb);
      v2f b2 = *(const v2f*)(W2 + k0 + kb);
      v2f b3 = *(const v2f*)(W3 + k0 + kb);
      acc0 = __builtin_amdgcn_wmma_f32_16x16x4_f32(false, a, false, b0, (short)0, acc0, false, false);
      acc1 = __builtin_amdgcn_wmma_f32_16x16x4_f32(false, a, false, b1, (short)0, acc1, false, false);
      acc2 = __builtin_amdgcn_wmma_f32_16x16x4_f32(false, a, false, b2, (short)0, acc2, false, false);
      acc3 = __builtin_amdgcn_wmma_f32_16x16x4_f32(false, a, false, b3, (short)0, acc3, false, false);
    }
    // mean + bias (exact DGL semantics: empty segment -> 0), stage to LDS
#pragma unroll
    for (int r = 0; r < 8; ++r) {
      int node = node0 + rhi + r;
      float c = cnt[node];
      float inv = 1.0f / fmaxf(c, 1.0f);
      hstage[wave][rhi + r][m]      = (acc0[r] + c * bmsg[m])      * inv;
      hstage[wave][rhi + r][m + 16] = (acc1[r] + c * bmsg[m + 16]) * inv;
      hstage[wave][rhi + r][m + 32] = (acc2[r] + c * bmsg[m + 32]) * inv;
      hstage[wave][rhi + r][m + 48] = (acc3[r] + c * bmsg[m + 48]) * inv;
    }
  }
  __syncthreads();

  if (active) {
    // ---- Stage 2: [16x128] @ W_apply^T[128x64], A = concat(nfeats, h_neigh) ----
    acc0 = (v8f){}; acc1 = (v8f){}; acc2 = (v8f){}; acc3 = (v8f){};
    const float* nrow = nfeats + (long)(node0 + m) * D;
    const float* hrow = &hstage[wave][m][0];
    const float* W0 = Wapp + (long)(m)      * DK;
    const float* W1 = Wapp + (long)(m + 16) * DK;
    const float* W2 = Wapp + (long)(m + 32) * DK;
    const float* W3 = Wapp + (long)(m + 48) * DK;
#pragma unroll 8
    for (int k0 = 0; k0 < DK; k0 += 4) {
      v2f a = (k0 < D) ? *(const v2f*)(nrow + k0 + kb)
                       : *(const v2f*)(hrow + (k0 - D) + kb);
      v2f b0 = *(const v2f*)(W0 + k0 + kb);
      v2f b1 = *(const v2f*)(W1 + k0 + kb);
      v2f b2 = *(const v2f*)(W2 + k0 + kb);
      v2f b3 = *(const v2f*)(W3 + k0 + kb);
      acc0 = __builtin_amdgcn_wmma_f32_16x16x4_f32(false, a, false, b0, (short)0, acc0, false, false);
      acc1 = __builtin_amdgcn_wmma_f32_16x16x4_f32(false, a, false, b1, (short)0, acc1, false, false);
      acc2 = __builtin_amdgcn_wmma_f32_16x16x4_f32(false, a, false, b2, (short)0, acc2, false, false);
      acc3 = __builtin_amdgcn_wmma_f32_16x16x4_f32(false, a, false, b3, (short)0, acc3, false, false);
    }
    // bias + relu + store (lanes 0-15 write 64B contiguous per row segment)
#pragma unroll
    for (int r = 0; r < 8; ++r) {
      float* orow = out + (long)(node0 + rhi + r) * D;
      orow[m]      = fmaxf(acc0[r] + bapp[m],      0.f);
      orow[m + 16] = fmaxf(acc1[r] + bapp[m + 16], 0.f);
      orow[m + 32] = fmaxf(acc2[r] + bapp[m + 32], 0.f);
      orow[m + 48] = fmaxf(acc3[r] + bapp[m + 48], 0.f);
    }
  }
}

extern "C" void kernel_launch(void* const* d_in, const int* in_sizes, int n_in,
                              void* d_out, int out_size, void* d_ws, size_t ws_size,
                              hipStream_t stream) {
  const float* nfeats = (const float*)d_in[0];
  const float* efeats = (const float*)d_in[1];
  const int*   src    = (const int*)d_in[2];
  const int*   dst    = (const int*)d_in[3];
  const float* Wmsg   = (const float*)d_in[4];
  const float* bmsg   = (const float*)d_in[5];
  const float* Wapp   = (const float*)d_in[6];
  const float* bapp   = (const float*)d_in[7];
  float* out = (float*)d_out;

  float* S   = (float*)d_ws;
  float* cnt = S + (size_t)NN * DK;

  // 1) zero accumulators (every call: launch must be self-contained/deterministic)
  const int n4 = NN * (DK + 1) / 4;  // 12.9M floats, divisible by 4
  zero_ws_kernel<<<(n4 + 255) / 256, 256, 0, stream>>>((float4*)d_ws, n4);

  // 2) edge scatter: 64 threads/edge
  const long ethreads = (long)NE * 64;
  edge_scatter_kernel<<<(int)((ethreads + 255) / 256), 256, 0, stream>>>(
      nfeats, efeats, src, dst, S, cnt);

  // 3) fused node-side double GEMM (WMMA fp32), 128 nodes per block
  const int tiles = NN / 16;                 // 6250
  node_wmma_kernel<<<(tiles + 7) / 8, 256, 0, stream>>>(
      nfeats, S, cnt, Wmsg, bmsg, Wapp, bapp, out);
}